// GPPT_72421738545551
// MI455X (gfx1250) — compile-verified
//
#include <hip/hip_runtime.h>

#define N_NODES 131072
#define NGRAPH  16384
#define IN_CH   1024
#define OUT_CH  512
#define NUM_CENTERS 16
#define MAX_TILES 1040   // <= B/16 + 15 ragged tiles

typedef __attribute__((ext_vector_type(2))) float v2f;
typedef __attribute__((ext_vector_type(8))) float v8f;
typedef __attribute__((ext_vector_type(4))) int   v4i;

#define AS1 __attribute__((address_space(1)))
#define AS3 __attribute__((address_space(3)))

#if defined(__gfx1250__) && __has_builtin(__builtin_amdgcn_global_load_async_to_lds_b128)
#define HAVE_ASYNC_LDS 1
#else
#define HAVE_ASYNC_LDS 0
#endif

__device__ __forceinline__ void wait_asynccnt0() {
#if __has_builtin(__builtin_amdgcn_s_wait_asynccnt)
    __builtin_amdgcn_s_wait_asynccnt(0);
#else
    asm volatile("s_wait_asynccnt 0" ::: "memory");
#endif
}

// ---------------- Kernel 1: gather ego rows + expert routing (argmax) -------
__global__ __launch_bounds__(256) void route_kernel(
    const float* __restrict__ gnn_x, const float* __restrict__ Wc,
    const int* __restrict__ batch, const int* __restrict__ ego_idx,
    int* __restrict__ center_idx, int* __restrict__ ego_rows)
{
    __shared__ float xs[IN_CH];
    __shared__ float part[NUM_CENTERS][16];
    __shared__ float score[NUM_CENTERS];
    __shared__ int   s_row;
    const int b = blockIdx.x, tid = threadIdx.x;

    if (tid == 0) {
        // searchsorted(batch, b) on sorted segment ids
        int lo = 0, hi = N_NODES;
        while (lo < hi) { int mid = (lo + hi) >> 1; if (batch[mid] < b) lo = mid + 1; else hi = mid; }
        int row = lo + ego_idx[b];
        s_row = row;
        ego_rows[b] = row;
    }
    __syncthreads();
    const int row = s_row;
    const float* __restrict__ xr = gnn_x + (size_t)row * IN_CH;
    for (int i = tid; i < IN_CH; i += 256) xs[i] = xr[i];
    __syncthreads();

    const int c = tid >> 4, sub = tid & 15;
    const float* __restrict__ wr = Wc + (size_t)c * IN_CH;
    float acc = 0.f;
    for (int k = sub; k < IN_CH; k += 16) acc += xs[k] * wr[k];
    part[c][sub] = acc;
    __syncthreads();
    if (tid < NUM_CENTERS) {
        float s = 0.f;
        #pragma unroll
        for (int j = 0; j < 16; ++j) s += part[tid][j];
        score[tid] = s;
    }
    __syncthreads();
    if (tid == 0) {
        int best = 0; float bs = score[0];
        #pragma unroll
        for (int cc = 1; cc < NUM_CENTERS; ++cc)
            if (score[cc] > bs) { bs = score[cc]; best = cc; }   // strict > keeps first index on ties
        center_idx[b] = best;
    }
}

// ---------------- Kernel 2: histogram + prefix + tile table -----------------
__global__ __launch_bounds__(256) void build_tiles_kernel(
    const int* __restrict__ center_idx,
    int* __restrict__ counts, int* __restrict__ offsets, int* __restrict__ fill,
    int* __restrict__ ntiles, int* __restrict__ tile_e,
    int* __restrict__ tile_start, int* __restrict__ tile_rows)
{
    __shared__ int lc[NUM_CENTERS];
    const int tid = threadIdx.x;
    if (tid < NUM_CENTERS) lc[tid] = 0;
    __syncthreads();
    for (int b = tid; b < NGRAPH; b += 256) atomicAdd(&lc[center_idx[b]], 1);
    __syncthreads();
    if (tid < NUM_CENTERS) { counts[tid] = lc[tid]; fill[tid] = 0; }
    __syncthreads();
    if (tid == 0) {
        int off = 0, t = 0;
        for (int e = 0; e < NUM_CENTERS; ++e) {
            offsets[e] = off;
            const int cnt = lc[e];
            for (int r = 0; r < cnt; r += 16) {
                tile_e[t]     = e;
                tile_start[t] = off + r;
                tile_rows[t]  = (cnt - r < 16) ? (cnt - r) : 16;
                ++t;
            }
            off += cnt;
        }
        offsets[NUM_CENTERS] = off;
        *ntiles = t;
    }
}

// ---------------- Kernel 3: scatter b's into per-expert buckets -------------
__global__ __launch_bounds__(256) void scatter_kernel(
    const int* __restrict__ center_idx, const int* __restrict__ offsets,
    int* __restrict__ fill, int* __restrict__ bucket_rows)
{
    const int b = blockIdx.x * 256 + threadIdx.x;
    if (b >= NGRAPH) return;
    const int e = center_idx[b];
    const int pos = atomicAdd(&fill[e], 1);
    bucket_rows[offsets[e] + pos] = b;
}

// ---------------- Kernel 4: grouped expert GEMM via V_WMMA_F32_16X16X4_F32 --
// Block = 128 threads (4 waves). Each block: one 16-row M-tile x 64 N columns.
// Each wave owns a 16x16 output tile; K=1024 in 64-wide LDS chunks,
// double-buffered via GLOBAL_LOAD_ASYNC_TO_LDS_B128 (ASYNCcnt) when available.
__global__ __launch_bounds__(128) void expert_gemm_kernel(
    const float* __restrict__ gnn_x, const float* __restrict__ W_experts,
    const int* __restrict__ ego_rows, const int* __restrict__ bucket_rows,
    const int* __restrict__ ntiles, const int* __restrict__ tile_e,
    const int* __restrict__ tile_start, const int* __restrict__ tile_rows,
    float* __restrict__ out)
{
    const int t = blockIdx.x;
    if (t >= *ntiles) return;                 // uniform branch: EXEC stays all-ones

#if HAVE_ASYNC_LDS
    __shared__ float As[2][16][68];           // [buf][m][k], pad 68 -> conflict-free frags
    __shared__ float Bs[2][64][68];           // [buf][n][k]
#else
    __shared__ float As[1][16][68];
    __shared__ float Bs[1][64][68];
#endif
    __shared__ int   s_grow[16];              // gnn_x row per m
    __shared__ int   s_b[16];                 // graph id per m

    const int tid = threadIdx.x;
    const int e     = tile_e[t];
    const int start = tile_start[t];
    const int nrows = tile_rows[t];

    if (tid < 16) {
        const int bi = bucket_rows[start + ((tid < nrows) ? tid : 0)];
        s_b[tid]    = bi;
        s_grow[tid] = ego_rows[bi];
    }
    __syncthreads();

    const int n0   = blockIdx.y * 64;
    const float* __restrict__ Wb = W_experts + (size_t)e * OUT_CH * IN_CH;

    const int wave  = tid >> 5, lane = tid & 31;
    const int n_sub = wave * 16;
    const int ml    = lane & 15;
    const int koff  = (lane < 16) ? 0 : 2;    // A/B 16x4 fp32 layout: hi half-wave holds K=2,3

    v8f acc = {};

#if HAVE_ASYNC_LDS
    // Per-lane issue pattern: A tile 16x64 f32 = 256 x 16B segments (2 per lane),
    // B tile 64x64 f32 = 1024 x 16B segments (8 per lane). Consecutive lanes cover
    // 256B contiguous global runs -> fully coalesced async bursts straight to LDS.
    const int m0   = tid >> 4;                // 0..7
    const int k4   = (tid & 15) * 4;          // 0,4,...,60
    const int rowA0 = s_grow[m0];
    const int rowA1 = s_grow[m0 + 8];

    auto issue_chunk = [&](int k0, int buf) {
        __builtin_amdgcn_global_load_async_to_lds_b128(
            (AS1 v4i*)(gnn_x + (size_t)rowA0 * IN_CH + k0 + k4),
            (AS3 v4i*)&As[buf][m0][k4], 0, 0);
        __builtin_amdgcn_global_load_async_to_lds_b128(
            (AS1 v4i*)(gnn_x + (size_t)rowA1 * IN_CH + k0 + k4),
            (AS3 v4i*)&As[buf][m0 + 8][k4], 0, 0);
        #pragma unroll
        for (int it = 0; it < 8; ++it) {
            const int s = tid + it * 128;
            const int n = s >> 4, kk4 = (s & 15) * 4;
            __builtin_amdgcn_global_load_async_to_lds_b128(
                (AS1 v4i*)(Wb + (size_t)(n0 + n) * IN_CH + k0 + kk4),
                (AS3 v4i*)&Bs[buf][n][kk4], 0, 0);
        }
    };

    issue_chunk(0, 0);
    for (int chunk = 0; chunk < IN_CH / 64; ++chunk) {
        const int buf = chunk & 1;
        wait_asynccnt0();                     // my async writes to LDS done
        __syncthreads();                      // everyone's done; prev buf fully consumed
        if (chunk + 1 < IN_CH / 64)
            issue_chunk((chunk + 1) * 64, buf ^ 1);   // overlap next load with compute

        #pragma unroll
        for (int kk = 0; kk < 64; kk += 4) {
            const int kb = kk + koff;
            v2f a, bf;
            a.x  = As[buf][ml][kb];            a.y  = As[buf][ml][kb + 1];
            bf.x = Bs[buf][n_sub + ml][kb];    bf.y = Bs[buf][n_sub + ml][kb + 1];
            acc = __builtin_amdgcn_wmma_f32_16x16x4_f32(
                false, a, false, bf, (short)0, acc, false, false);
        }
    }
#else
    for (int k0 = 0; k0 < IN_CH; k0 += 64) {
        for (int idx = tid; idx < 16 * 64; idx += 128) {
            const int m = idx >> 6, k = idx & 63;
            As[0][m][k] = gnn_x[(size_t)s_grow[m] * IN_CH + k0 + k];
        }
        for (int idx = tid; idx < 64 * 64; idx += 128) {
            const int n = idx >> 6, k = idx & 63;
            Bs[0][n][k] = Wb[(size_t)(n0 + n) * IN_CH + k0 + k];
        }
        __syncthreads();
        #pragma unroll
        for (int kk = 0; kk < 64; kk += 4) {
            const int kb = kk + koff;
            v2f a, bf;
            a.x  = As[0][ml][kb];            a.y  = As[0][ml][kb + 1];
            bf.x = Bs[0][n_sub + ml][kb];    bf.y = Bs[0][n_sub + ml][kb + 1];
            acc = __builtin_amdgcn_wmma_f32_16x16x4_f32(
                false, a, false, bf, (short)0, acc, false, false);
        }
        __syncthreads();
    }
#endif

    // C/D layout: VGPR v, lanes 0-15 -> M=v, N=lane; lanes 16-31 -> M=v+8, N=lane-16
    const int nc = n0 + n_sub + ml;
    #pragma unroll
    for (int v = 0; v < 8; ++v) {
        const int m = v + ((lane < 16) ? 0 : 8);
        if (m < nrows) out[(size_t)s_b[m] * OUT_CH + nc] = acc[v];
    }
}

// ---------------------------------------------------------------------------
extern "C" void kernel_launch(void* const* d_in, const int* in_sizes, int n_in,
                              void* d_out, int out_size, void* d_ws, size_t ws_size,
                              hipStream_t stream) {
    const float* gnn_x     = (const float*)d_in[0];
    const float* Wc        = (const float*)d_in[1];
    const float* W_experts = (const float*)d_in[2];
    const int*   batch     = (const int*)d_in[3];
    const int*   ego_idx   = (const int*)d_in[4];
    float*       out       = (float*)d_out;

    int* ws          = (int*)d_ws;
    int* center_idx  = ws;                        // [B]
    int* ego_rows    = ws + NGRAPH;               // [B]
    int* bucket_rows = ws + 2 * NGRAPH;           // [B]
    int* base        = ws + 3 * NGRAPH;
    int* counts      = base;                      // [16]
    int* offsets     = base + 16;                 // [17]
    int* fill        = base + 33;                 // [16]
    int* ntiles      = base + 49;                 // [1]
    int* tile_e      = base + 64;                 // [MAX_TILES]
    int* tile_start  = tile_e + MAX_TILES;        // [MAX_TILES]
    int* tile_rows   = tile_start + MAX_TILES;    // [MAX_TILES]

    route_kernel<<<NGRAPH, 256, 0, stream>>>(gnn_x, Wc, batch, ego_idx,
                                             center_idx, ego_rows);
    build_tiles_kernel<<<1, 256, 0, stream>>>(center_idx, counts, offsets, fill,
                                              ntiles, tile_e, tile_start, tile_rows);
    scatter_kernel<<<(NGRAPH + 255) / 256, 256, 0, stream>>>(center_idx, offsets,
                                                             fill, bucket_rows);
    expert_gemm_kernel<<<dim3(MAX_TILES, OUT_CH / 64), 128, 0, stream>>>(
        gnn_x, W_experts, ego_rows, bucket_rows,
        ntiles, tile_e, tile_start, tile_rows, out);
}